// RNNsBlock_1632087572932
// MI455X (gfx1250) — compile-verified
//
#include <hip/hip_runtime.h>
#include <cmath>

// Problem constants (match reference).
constexpr int T = 1024;
constexpr int B = 64;
constexpr int E = 512;
constexpr int H = 512;
constexpr int L = 2;
constexpr int M_PROJ = T * B;          // 65536 rows in the projection GEMM
constexpr int REC_WGS = 16;            // persistent workgroups in recurrence
constexpr int REC_NCOLS = H / REC_WGS; // 32 output columns per WG

typedef float v2f __attribute__((ext_vector_type(2)));
typedef float v8f __attribute__((ext_vector_type(8)));
typedef unsigned int u32x4 __attribute__((ext_vector_type(4)));
typedef int i32x4 __attribute__((ext_vector_type(4)));
typedef int i32x8 __attribute__((ext_vector_type(8)));

// ---------------------------------------------------------------------------
// V_WMMA_F32_16X16X4_F32 fragment helpers (wave32).
// A (16x4, MxK):  lane<16: M=lane, holds K=k+0,k+1 ; lane>=16: M=lane-16, K=k+2,k+3
// B (4x16, KxN):  lane<16: N=lane, holds K=k+0,k+1 ; lane>=16: N=lane-16, K=k+2,k+3
// C/D (16x16):    VGPR i: lanes 0-15 -> M=i, lanes 16-31 -> M=8+i ; N=lane%16
// ---------------------------------------------------------------------------
__device__ __forceinline__ v2f load_a_frag(const float* __restrict__ A, int lda,
                                           int m_base, int k, int lane) {
  const int half = lane >> 4;
  const int ml   = lane & 15;
  const float* p = A + (size_t)(m_base + ml) * lda + (k + 2 * half);
  return *reinterpret_cast<const v2f*>(p);   // 8B aligned: k%4==0, lda%2==0
}

__device__ __forceinline__ v2f load_b_frag(const float* __restrict__ Bm, int ldb,
                                           int k, int n_base, int lane) {
  const int half = lane >> 4;
  const int nl   = lane & 15;
  const float* p = Bm + (size_t)(k + 2 * half) * ldb + (n_base + nl);
  v2f r;
  r.x = p[0];        // row k+2*half
  r.y = p[ldb];      // row k+2*half+1
  return r;
}

__device__ __forceinline__ v8f wmma_f32(v2f a, v2f b, v8f c) {
  // 8 args: (neg_a, A, neg_b, B, c_mod, C, reuse_a, reuse_b)
  return __builtin_amdgcn_wmma_f32_16x16x4_f32(false, a, false, b, (short)0, c,
                                               false, false);
}

// ---------------------------------------------------------------------------
// Kernel 1: xw = X @ Wx + bias.   X:[M,E] row-major, Wx:[E,H] row-major.
// Workgroup tile 128(M) x 64(N): 8 waves as 4x2, each wave 32x32 (2x2 WMMA tiles).
// grid = (M/128, H/64)
// ---------------------------------------------------------------------------
__global__ __launch_bounds__(256) void rnn_proj_kernel(
    const float* __restrict__ X, const float* __restrict__ Wx,
    const float* __restrict__ bias, float* __restrict__ Out)
{
  const int lane = threadIdx.x & 31;
  const int wave = threadIdx.x >> 5;
  const int m0 = blockIdx.x * 128 + (wave & 3) * 32;
  const int n0 = blockIdx.y * 64  + (wave >> 2) * 32;

  v8f acc[2][2] = {};
  for (int k = 0; k < E; k += 4) {
    v2f a0 = load_a_frag(X, E, m0,      k, lane);
    v2f a1 = load_a_frag(X, E, m0 + 16, k, lane);
    v2f b0 = load_b_frag(Wx, H, k, n0,      lane);
    v2f b1 = load_b_frag(Wx, H, k, n0 + 16, lane);
    acc[0][0] = wmma_f32(a0, b0, acc[0][0]);
    acc[0][1] = wmma_f32(a0, b1, acc[0][1]);
    acc[1][0] = wmma_f32(a1, b0, acc[1][0]);
    acc[1][1] = wmma_f32(a1, b1, acc[1][1]);
  }

  const int half = lane >> 4;
  const int nl   = lane & 15;
  #pragma unroll
  for (int mi = 0; mi < 2; ++mi)
    #pragma unroll
    for (int ni = 0; ni < 2; ++ni) {
      const int col = n0 + ni * 16 + nl;
      const float bv = bias[col];
      #pragma unroll
      for (int i = 0; i < 8; ++i) {
        const int row = m0 + mi * 16 + 8 * half + i;
        Out[(size_t)row * H + col] = acc[mi][ni][i] + bv;
      }
    }
}

// ---------------------------------------------------------------------------
// Kernel 2: persistent recurrence, in place over YS[T,B,H] which holds xw on
// entry:  YS[t] = tanh(YS[t] + YS[t-1] @ Wh)   (YS[-1] := h0)
//
// 16 persistent WGs, each owns 32 output columns. The WG's Wh slice
// (512 x 32 = 64 KB) is staged ONCE into LDS via the Tensor Data Mover
// (one 2D-tile descriptor, TENSORcnt path) and reused for all 1024 steps.
// B-fragments come from conflict-free ds loads. Per step: 64x32 tile, K=512,
// 8 waves as 4(M)x2(N). Grid-wide release/acquire barrier per step.
// ---------------------------------------------------------------------------
__global__ __launch_bounds__(256) void rnn_rec_kernel(
    const float* __restrict__ Wh,    // [H,H] row-major (k, n)
    const float* __restrict__ h0,    // [B,H] initial state
    float* __restrict__ YS,          // [T,B,H] xw -> ys (in place)
    float* __restrict__ h_last,     // [B,H] final state out
    int* barrier_cnt)
{
  __shared__ float whs[H * REC_NCOLS];           // 64 KB: Wh[k][n0 .. n0+32)

  const int tid   = threadIdx.x;
  const int ncol0 = blockIdx.x * REC_NCOLS;

  // ---- TDM: one descriptor-driven DMA of the 512x32 f32 tile ------------
  if (tid == 0) {
    const unsigned long long ga =
        (unsigned long long)(uintptr_t)(Wh + ncol0);    // tile start (bytes)
    const unsigned int lds_addr = (unsigned int)(uintptr_t)&whs[0];

    u32x4 g0;
    g0[0] = 1u;                                         // count=1, user mode
    g0[1] = lds_addr;                                   // lds_addr [63:32]
    g0[2] = (unsigned int)(ga & 0xFFFFFFFFull);         // global_addr lo
    g0[3] = (unsigned int)((ga >> 32) & 0x1FFFFFFull)   // global_addr [56:32]
            | (2u << 30);                               // type=2 ("image")

    i32x8 g1;
    g1[0] = (2 << 16);            // workgroup_mask=0, data_size=2 (4B)
    g1[1] = (H & 0xFFFF) << 16;   // tensor_dim0[15:0] in bits [63:48]
    g1[2] = (H & 0xFFFF) << 16;   // dim0 hi=0 (bits 79:64); tensor_dim1 lo in [95:80]
    g1[3] = (REC_NCOLS) << 16;    // tensor_dim1 hi=0; tile_dim0=32 in [127:112]
    g1[4] = H;                    // tile_dim1=512 in [143:128]; tile_dim2=0
    g1[5] = H;                    // tensor_dim0_stride[31:0] = 512
    g1[6] = 0;                    // stride hi=0; tensor_dim1_stride lo=0 (unused, 2D)
    g1[7] = 0;

    i32x4 g2 = {0, 0, 0, 0};      // 2D: dims 2/3 unused
    i32x4 g3 = {0, 0, 0, 0};
    i32x8 g4 = {0, 0, 0, 0, 0, 0, 0, 0};   // 6-arg toolchain form: extra group

    __builtin_amdgcn_tensor_load_to_lds(g0, g1, g2, g3, g4, 0);
    __builtin_amdgcn_s_wait_tensorcnt(0);
  }
  __syncthreads();   // publish LDS tile to all waves

  const int lane = tid & 31;
  const int wave = tid >> 5;
  const int m0   = (wave & 3) * 16;              // 4 waves over 64 batch rows
  const int nw   = (wave >> 2) * 16;             // 2 waves over 32 columns
  const int half = lane >> 4;
  const int nl   = lane & 15;
  const int gcol = ncol0 + nw + nl;              // global output column

  for (int t = 0; t < T; ++t) {
    const float* hprev = (t == 0) ? h0 : (YS + (size_t)(t - 1) * B * H);

    // Pull next step's xw lines toward the WGP while we crunch this step.
    if (t + 1 < T) {
      __builtin_prefetch(YS + (size_t)(t + 1) * B * H +
                             (size_t)(m0 + 8 * half) * H + gcol, 0, 3);
    }

    v8f acc = {};
    for (int k = 0; k < H; k += 4) {
      v2f a = load_a_frag(hprev, H, m0, k, lane);
      v2f b;                                      // from LDS (ds loads)
      b.x = whs[(k + 2 * half)     * REC_NCOLS + nw + nl];
      b.y = whs[(k + 2 * half + 1) * REC_NCOLS + nw + nl];
      acc = wmma_f32(a, b, acc);
    }

    float* y = YS + (size_t)t * B * H;
    #pragma unroll
    for (int i = 0; i < 8; ++i) {
      const int row = m0 + 8 * half + i;
      const size_t idx = (size_t)row * H + gcol;
      const float v = tanhf(y[idx] + acc[i]);
      y[idx] = v;
      if (t == T - 1) h_last[idx] = v;
    }

    // ---- grid-wide barrier: everyone's YS[t] visible before step t+1 ----
    __threadfence();          // make this thread's global writes visible
    __syncthreads();          // whole WG done writing
    if (tid == 0) {
      __hip_atomic_fetch_add(barrier_cnt, 1, __ATOMIC_RELEASE,
                             __HIP_MEMORY_SCOPE_AGENT);
      const int target = REC_WGS * (t + 1);
      while (__hip_atomic_load(barrier_cnt, __ATOMIC_ACQUIRE,
                               __HIP_MEMORY_SCOPE_AGENT) < target) {
        __builtin_amdgcn_s_sleep(1);
      }
    }
    __syncthreads();          // rest of WG waits for thread0's acquire
  }
}

// ---------------------------------------------------------------------------
// Host-side launch. d_in order: input [T,B,E], states [L,B,H], W_xh [L,H,H],
// W_hh [L,H,H], b_h [L,H].  d_out: ys [L,T,B,H] then last_states [L,B,H].
// ---------------------------------------------------------------------------
extern "C" void kernel_launch(void* const* d_in, const int* in_sizes, int n_in,
                              void* d_out, int out_size, void* d_ws, size_t ws_size,
                              hipStream_t stream) {
  (void)in_sizes; (void)n_in; (void)out_size; (void)ws_size;

  const float* x      = (const float*)d_in[0];
  const float* states = (const float*)d_in[1];
  const float* W_xh   = (const float*)d_in[2];
  const float* W_hh   = (const float*)d_in[3];
  const float* b_h    = (const float*)d_in[4];

  float* out = (float*)d_out;
  float* ys0 = out;                                   // layer 0 outputs
  float* ys1 = out + (size_t)T * B * H;               // layer 1 outputs
  float* hl  = out + (size_t)L * T * B * H;           // [L,B,H] last states

  int* counters = (int*)d_ws;                         // two barrier counters
  (void)hipMemsetAsync(d_ws, 0, 2 * sizeof(int), stream);

  const dim3 pgrid(M_PROJ / 128, H / 64);             // 512 x 8
  const dim3 blk(256);

  // Layer 0
  rnn_proj_kernel<<<pgrid, blk, 0, stream>>>(x, W_xh, b_h, ys0);
  rnn_rec_kernel<<<dim3(REC_WGS), blk, 0, stream>>>(W_hh, states, ys0,
                                                    hl, counters + 0);
  // Layer 1 (input = layer 0 outputs)
  rnn_proj_kernel<<<pgrid, blk, 0, stream>>>(ys0, W_xh + (size_t)H * E,
                                             b_h + H, ys1);
  rnn_rec_kernel<<<dim3(REC_WGS), blk, 0, stream>>>(W_hh + (size_t)H * H,
                                                    states + (size_t)B * H, ys1,
                                                    hl + (size_t)B * H,
                                                    counters + 1);
}